// Word2VecCBOW_15350213116310
// MI455X (gfx1250) — compile-verified
//
#include <hip/hip_runtime.h>
#include <hip/hip_bf16.h>
#include <math.h>

// Word2Vec CBOW negative-sampling loss for MI455X (gfx1250, wave32).
// Gather-bound; per-row dot products run on v_wmma_f32_16x16x4_f32 chains,
// candidate loop unrolled 2x for independent WMMA chains + deeper load pipeline.

typedef float v2f __attribute__((ext_vector_type(2)));
typedef float v8f __attribute__((ext_vector_type(8)));

#define DIM      128   // embedding dim
#define CWIN     10    // context window
#define KNEG     10    // negative samples
#define WAVES_PB 4     // waves per block (each wave owns a 16-row tile)

__device__ __forceinline__ float log_sigmoid(float x) {
    // stable: min(x,0) - log1p(exp(-|x|))
    return fminf(x, 0.0f) - log1pf(expf(-fabsf(x)));
}

__global__ __launch_bounds__(32 * WAVES_PB)
void cbow_wmma_kernel(const int* __restrict__ context,     // (B, CWIN)
                      const int* __restrict__ target,      // (B,)
                      const int* __restrict__ neg,         // (B, KNEG)
                      const float* __restrict__ Win,       // (V, DIM)
                      const float* __restrict__ Wout,      // (V, DIM)
                      float* __restrict__ partial)         // (gridDim.x,)
{
    __shared__ float diag[WAVES_PB][2][16][17];  // wave-private C-tile spill, 2 slots (+pad)
    __shared__ float wsum[WAVES_PB];

    const int tid  = threadIdx.x;
    const int wave = tid >> 5;        // wave32
    const int lane = tid & 31;
    const int half = lane >> 4;       // 0: K pair {0,1}, 1: K pair {2,3} within 4-wide step
    const int n    = lane & 15;       // row-in-tile (A: M index; B: N index)

    const int tile = blockIdx.x * WAVES_PB + wave;  // 16 batch rows per wave
    const long r   = (long)tile * 16 + n;           // this lane's batch row

    // ---------------- Build H fragments in WMMA A-matrix layout ----------------
    // A (16x4 f32): lanes 0-15 hold M=0..15 with VGPR0=K0,VGPR1=K1; lanes 16-31 K2,K3.
    // h2[j] = { h[n][4j + 2*half], h[n][4j + 2*half + 1] }, j = 0..31 covers D=128.
    v2f h2[DIM / 4];
#pragma unroll
    for (int j = 0; j < DIM / 4; ++j) { h2[j].x = 0.0f; h2[j].y = 0.0f; }

#pragma unroll
    for (int c = 0; c < CWIN; ++c) {
        const int idx = context[r * CWIN + c];
        const float* row = Win + (long)idx * DIM + 2 * half;
#pragma unroll
        for (int j = 0; j < DIM / 4; ++j) {
            const v2f e = *reinterpret_cast<const v2f*>(row + 4 * j);  // global_load_b64
            h2[j] += e;
        }
    }
    const float invC = 1.0f / (float)CWIN;
#pragma unroll
    for (int j = 0; j < DIM / 4; ++j) h2[j] *= invC;

    // ---------------- Score candidates: k=0 positive, k=1..KNEG negative --------
    // Processed in pairs -> two independent wmma accumulation chains per pass.
    float rowAcc = 0.0f;  // pos_loss + neg_loss for row n (valid on lanes 0-15)

    for (int kp = 0; kp < (KNEG + 1) / 2; ++kp) {      // pairs (0,1),(2,3)...(8,9)
        const int ka = 2 * kp;
        const int kb = ka + 1;
        const int ta = (ka == 0) ? target[r] : neg[r * KNEG + (ka - 1)];
        const int tb = neg[r * KNEG + (kb - 1)];
        const float* ea = Wout + (long)ta * DIM + 2 * half;
        const float* eb = Wout + (long)tb * DIM + 2 * half;

        v8f accA = {0.f, 0.f, 0.f, 0.f, 0.f, 0.f, 0.f, 0.f};
        v8f accB = {0.f, 0.f, 0.f, 0.f, 0.f, 0.f, 0.f, 0.f};
#pragma unroll
        for (int j = 0; j < DIM / 4; ++j) {
            const v2f ba = *reinterpret_cast<const v2f*>(ea + 4 * j);
            const v2f bb = *reinterpret_cast<const v2f*>(eb + 4 * j);
            accA = __builtin_amdgcn_wmma_f32_16x16x4_f32(
                false, h2[j], false, ba, (short)0, accA, false, false);
            accB = __builtin_amdgcn_wmma_f32_16x16x4_f32(
                false, h2[j], false, bb, (short)0, accB, false, false);
        }

        // Diagonal extraction: C/D layout lane<16: M=v,N=lane; lane>=16: M=v+8,N=lane-16.
#pragma unroll
        for (int v = 0; v < 8; ++v) {
            diag[wave][0][v + 8 * half][n] = accA[v];
            diag[wave][1][v + 8 * half][n] = accB[v];
        }
        __builtin_amdgcn_wave_barrier();     // wave-private LDS; DS in-order per wave
        const float sA = diag[wave][0][n][n];
        const float sB = diag[wave][1][n][n];
        __builtin_amdgcn_wave_barrier();

        const float cA = (ka == 0) ? log_sigmoid(sA) : log_sigmoid(-sA);
        const float cB = log_sigmoid(-sB);
        rowAcc += (lane < 16) ? (cA + cB) : 0.0f;  // lanes 16-31 duplicate rows -> mask
    }

    {   // tail candidate k = KNEG (last negative)
        const int t = neg[r * KNEG + (KNEG - 1)];
        const float* erow = Wout + (long)t * DIM + 2 * half;
        v8f acc = {0.f, 0.f, 0.f, 0.f, 0.f, 0.f, 0.f, 0.f};
#pragma unroll
        for (int j = 0; j < DIM / 4; ++j) {
            const v2f b2 = *reinterpret_cast<const v2f*>(erow + 4 * j);
            acc = __builtin_amdgcn_wmma_f32_16x16x4_f32(
                false, h2[j], false, b2, (short)0, acc, false, false);
        }
#pragma unroll
        for (int v = 0; v < 8; ++v)
            diag[wave][0][v + 8 * half][n] = acc[v];
        __builtin_amdgcn_wave_barrier();
        const float s = diag[wave][0][n][n];
        __builtin_amdgcn_wave_barrier();
        rowAcc += (lane < 16) ? log_sigmoid(-s) : 0.0f;
    }

    // ---------------- Wave reduction over 16 rows -> block partial ---------------
#pragma unroll
    for (int off = 16; off >= 1; off >>= 1)
        rowAcc += __shfl_xor(rowAcc, off, 32);
    if (lane == 0) wsum[wave] = rowAcc;
    __syncthreads();
    if (tid == 0) {
        float s = 0.0f;
#pragma unroll
        for (int w = 0; w < WAVES_PB; ++w) s += wsum[w];
        partial[blockIdx.x] = s;
    }
}

// Deterministic final reduction: -(sum of log-probs) / B
__global__ __launch_bounds__(256)
void cbow_reduce_kernel(const float* __restrict__ partial, float* __restrict__ out,
                        int nPartials, float invB)
{
    __shared__ float sm[256];
    const int t = threadIdx.x;
    float v = 0.0f;
    for (int i = t; i < nPartials; i += 256) v += partial[i];  // fixed order per thread
    sm[t] = v;
    __syncthreads();
#pragma unroll
    for (int s = 128; s > 0; s >>= 1) {
        if (t < s) sm[t] += sm[t + s];
        __syncthreads();
    }
    if (t == 0) out[0] = -sm[0] * invB;
}

extern "C" void kernel_launch(void* const* d_in, const int* in_sizes, int n_in,
                              void* d_out, int out_size, void* d_ws, size_t ws_size,
                              hipStream_t stream) {
    const int*   context = (const int*)  d_in[0];   // (B, C)
    const int*   target  = (const int*)  d_in[1];   // (B,)
    const int*   neg     = (const int*)  d_in[2];   // (B, K)
    const float* Win     = (const float*)d_in[3];   // (V, D)
    const float* Wout    = (const float*)d_in[4];   // (V, D)
    float*       out     = (float*)d_out;
    float*       partial = (float*)d_ws;

    const int B = in_sizes[1];                      // 16384
    const int rowsPerBlock = 16 * WAVES_PB;         // 64
    const int nBlocks = B / rowsPerBlock;           // 256 (B divisible by 64)

    cbow_wmma_kernel<<<nBlocks, 32 * WAVES_PB, 0, stream>>>(
        context, target, neg, Win, Wout, partial);
    cbow_reduce_kernel<<<1, 256, 0, stream>>>(partial, out, nBlocks, 1.0f / (float)B);
}